// GNN_17669495455821
// MI455X (gfx1250) — compile-verified
//
#include <hip/hip_runtime.h>
#include <hip/hip_bf16.h>
#include <math.h>
#include <stdint.h>

// Problem constants (from reference)
#define B_  16
#define N_  1024
#define D_  512
#define K_  16
#define KP1 17
#define NEG_SLOPE 0.2f

typedef __attribute__((ext_vector_type(2))) float v2f;
typedef __attribute__((ext_vector_type(8))) float v8f;

__device__ __forceinline__ float wave_reduce_max(float v) {
#pragma unroll
  for (int off = 16; off > 0; off >>= 1)
    v = fmaxf(v, __shfl_xor(v, off, 32));
  return v;
}
__device__ __forceinline__ float wave_reduce_sum(float v) {
#pragma unroll
  for (int off = 16; off > 0; off >>= 1)
    v += __shfl_xor(v, off, 32);
  return v;
}

// ---------------------------------------------------------------------------
// NormalizeScale: h = (x - mean(x,axis=0)) * 0.999999 / max|x - mean|
// grid = B, block = D (512 threads, one per feature column)
// ---------------------------------------------------------------------------
__global__ void __launch_bounds__(D_) normalize_kernel(const float* __restrict__ x,
                                                       float* __restrict__ h) {
  __shared__ float red[D_];
  __shared__ float s_scale;
  const int b = blockIdx.x;
  const int d = threadIdx.x;
  const float* xb = x + (size_t)b * N_ * D_;
  float sum = 0.f;
  for (int n = 0; n < N_; ++n) sum += xb[(size_t)n * D_ + d];
  const float mean = sum * (1.0f / (float)N_);
  float mx = 0.f;
  for (int n = 0; n < N_; ++n) mx = fmaxf(mx, fabsf(xb[(size_t)n * D_ + d] - mean));
  red[d] = mx;
  __syncthreads();
  for (int s = D_ / 2; s > 0; s >>= 1) {
    if (d < s) red[d] = fmaxf(red[d], red[d + s]);
    __syncthreads();
  }
  if (d == 0) s_scale = 0.999999f / red[0];
  __syncthreads();
  const float scale = s_scale;
  float* hb = h + (size_t)b * N_ * D_;
  for (int n = 0; n < N_; ++n)
    hb[(size_t)n * D_ + d] = (xb[(size_t)n * D_ + d] - mean) * scale;
}

// ---------------------------------------------------------------------------
// Squared norms of original positions: sq[b*N+n] = sum_d x^2
// grid = B*N/8 blocks of 256 threads (wave32 per node)
// ---------------------------------------------------------------------------
__global__ void __launch_bounds__(256) sqnorm_kernel(const float* __restrict__ x,
                                                     float* __restrict__ sq) {
  const int wave = threadIdx.x >> 5, lane = threadIdx.x & 31;
  const int node = blockIdx.x * 8 + wave;   // flat over B*N
  const float* row = x + (size_t)node * D_;
  float s = 0.f;
#pragma unroll
  for (int t = 0; t < D_ / 32; ++t) {
    float v = row[lane + 32 * t];
    s += v * v;
  }
  s = wave_reduce_sum(s);
  if (lane == 0) sq[node] = s;
}

// ---------------------------------------------------------------------------
// KNN: Gram via V_WMMA_F32_16X16X4_F32, streamed top-16 selection per row.
// grid = (N/16, B), block = 32 (one wave per 16-row block)
// ---------------------------------------------------------------------------
__global__ void __launch_bounds__(32) knn_kernel(const float* __restrict__ x,
                                                 const float* __restrict__ sq,
                                                 int* __restrict__ nbr) {
  __shared__ float tile[16 * 16];
  const int b = blockIdx.y;
  const int row0 = blockIdx.x * 16;
  const int lane = threadIdx.x;
  const int half = lane >> 4, l16 = lane & 15;
  const float* xb  = x  + (size_t)b * N_ * D_;
  const float* sqb = sq + (size_t)b * N_;

  float bestd[16];
  int   besti[16];
#pragma unroll
  for (int j = 0; j < 16; ++j) { bestd[j] = INFINITY; besti[j] = 0; }

  const float* arow = xb + (size_t)(row0 + l16) * D_ + 2 * half;  // A frag base

  for (int ct = 0; ct < N_ / 16; ++ct) {
    const int col0 = ct * 16;
    const float* brow = xb + (size_t)(col0 + l16) * D_ + 2 * half; // B = X^T frag
    v8f c = {};
    for (int k = 0; k < D_; k += 4) {
      if ((k & 127) == 0) {
        __builtin_prefetch(brow + k + 128, 0, 1);
      }
      v2f a  = *(const v2f*)(arow + k);
      v2f bb = *(const v2f*)(brow + k);
      c = __builtin_amdgcn_wmma_f32_16x16x4_f32(false, a, false, bb,
                                                (short)0, c, false, false);
    }
    const float sqc = sqb[col0 + l16];
#pragma unroll
    for (int v = 0; v < 8; ++v) {
      const int m = v + 8 * half;              // C layout: M = v + 8*(lane>=16)
      float d2 = sqb[row0 + m] + sqc - 2.0f * c[v];
      if (row0 + m == col0 + l16) d2 = INFINITY;  // exclude self
      tile[m * 16 + l16] = d2;
    }
    __syncthreads();
    if (lane < 16) {
#pragma unroll
      for (int j = 0; j < 16; ++j) {
        const float d = tile[lane * 16 + j];
        if (d < bestd[15]) {
          bestd[15] = d; besti[15] = col0 + j;
#pragma unroll
          for (int s = 15; s > 0; --s) {     // one bubble pass keeps list sorted
            if (bestd[s] < bestd[s - 1]) {
              float td = bestd[s]; bestd[s] = bestd[s - 1]; bestd[s - 1] = td;
              int   ti = besti[s]; besti[s] = besti[s - 1]; besti[s - 1] = ti;
            }
          }
        }
      }
    }
    __syncthreads();
  }
  if (lane < 16) {
    const int grow = row0 + lane;
    int* out = nbr + ((size_t)b * N_ + grow) * KP1;
#pragma unroll
    for (int j = 0; j < 16; ++j) out[j] = besti[j];
    out[16] = grow;                           // self loop appended
  }
}

// ---------------------------------------------------------------------------
// hp = h @ W per graph. 4-wave workgroup computes a 64x64 output block.
// W k-slabs (32x64) are double-buffered in LDS via async global->LDS DMA
// (GLOBAL_LOAD_ASYNC_TO_LDS_B128 + s_wait_asynccnt), each wave computes a
// 16x64 strip with 4 f32-WMMA accumulators sharing one A fragment.
// grid = (N/64, D/64, B), block = 128
// ---------------------------------------------------------------------------
#define KSLAB   32
#define WROWPAD 72   // LDS row stride in floats (banks disjoint for the 2 halves)

__global__ void __launch_bounds__(128) gemm_kernel(const float* __restrict__ A,
                                                   const float* __restrict__ W,
                                                   float* __restrict__ Cout) {
  __shared__ float wst[2][KSLAB * WROWPAD];
  const int b = blockIdx.z;
  const int wave = threadIdx.x >> 5;
  const int row0 = blockIdx.x * 64 + wave * 16;
  const int col0 = blockIdx.y * 64;
  const int lane = threadIdx.x & 31;
  const int half = lane >> 4, l16 = lane & 15;
  const int tid = threadIdx.x;

  const float* Ab = A + (size_t)b * N_ * D_;
  const float* arow = Ab + (size_t)(row0 + l16) * D_ + 2 * half;

  // Staging assignment: thread tid owns slab row sr, 16-float chunk sc,
  // and issues 4 B128 async transfers (16 floats) per slab.
  const int sr = tid >> 2;            // 0..31
  const int sc = (tid & 3) * 16;      // 0,16,32,48
  const uint32_t lds0 =
      (uint32_t)(uintptr_t)&wst[0][sr * WROWPAD + sc];
  const uint32_t lds1 =
      (uint32_t)(uintptr_t)&wst[1][sr * WROWPAD + sc];
  const float* wsrc = W + (size_t)sr * D_ + col0 + sc;

  v8f c0 = {}, c1 = {}, c2 = {}, c3 = {};

#define ISSUE_SLAB(k0, ldsb)                                                   \
  {                                                                            \
    uint64_t ga = (uint64_t)(uintptr_t)(wsrc + (size_t)(k0)*D_);               \
    uint32_t la = (ldsb);                                                      \
    _Pragma("unroll")                                                          \
    for (int q = 0; q < 4; ++q) {                                              \
      uint64_t gq = ga + (uint64_t)(q * 16);                                   \
      uint32_t lq = la + (uint32_t)(q * 16);                                   \
      asm volatile("global_load_async_to_lds_b128 %0, %1, off"                 \
                   :: "v"(lq), "v"(gq) : "memory");                            \
    }                                                                          \
  }

  ISSUE_SLAB(0, lds0)
  asm volatile("s_wait_asynccnt 0x0" ::: "memory");
  __syncthreads();

  for (int s = 0; s < D_ / KSLAB; ++s) {
    const int buf = s & 1;
    if (s + 1 < D_ / KSLAB) {
      if (buf) { ISSUE_SLAB((s + 1) * KSLAB, lds0) }
      else     { ISSUE_SLAB((s + 1) * KSLAB, lds1) }
    }
    const float* wsl = &wst[buf][0];
    const float* ak = arow + s * KSLAB;
#pragma unroll
    for (int k = 0; k < KSLAB; k += 4) {
      v2f a = *(const v2f*)(ak + k);
      const int kk = k + 2 * half;
      const float* r0 = wsl + kk * WROWPAD + l16;
      const float* r1 = wsl + (kk + 1) * WROWPAD + l16;
      v2f b0, b1, b2, b3;
      b0.x = r0[0];  b0.y = r1[0];
      b1.x = r0[16]; b1.y = r1[16];
      b2.x = r0[32]; b2.y = r1[32];
      b3.x = r0[48]; b3.y = r1[48];
      c0 = __builtin_amdgcn_wmma_f32_16x16x4_f32(false, a, false, b0,
                                                 (short)0, c0, false, false);
      c1 = __builtin_amdgcn_wmma_f32_16x16x4_f32(false, a, false, b1,
                                                 (short)0, c1, false, false);
      c2 = __builtin_amdgcn_wmma_f32_16x16x4_f32(false, a, false, b2,
                                                 (short)0, c2, false, false);
      c3 = __builtin_amdgcn_wmma_f32_16x16x4_f32(false, a, false, b3,
                                                 (short)0, c3, false, false);
    }
    asm volatile("s_wait_asynccnt 0x0" ::: "memory");
    __syncthreads();
  }
#undef ISSUE_SLAB

  float* Cb = Cout + (size_t)b * N_ * D_;
#pragma unroll
  for (int v = 0; v < 8; ++v) {
    const int m = v + 8 * half;
    float* dst = Cb + (size_t)(row0 + m) * D_ + col0 + l16;
    dst[0]  = c0[v];
    dst[16] = c1[v];
    dst[32] = c2[v];
    dst[48] = c3[v];
  }
}

// ---------------------------------------------------------------------------
// a_s = hp @ asrc ; a_d = hp @ adst  (wave per node)
// ---------------------------------------------------------------------------
__global__ void __launch_bounds__(256) dots_kernel(const float* __restrict__ hp,
                                                   const float* __restrict__ asrc,
                                                   const float* __restrict__ adst,
                                                   float* __restrict__ a_s,
                                                   float* __restrict__ a_d) {
  const int wave = threadIdx.x >> 5, lane = threadIdx.x & 31;
  const int node = blockIdx.x * 8 + wave;
  const float* row = hp + (size_t)node * D_;
  float s1 = 0.f, s2 = 0.f;
#pragma unroll
  for (int t = 0; t < D_ / 32; ++t) {
    const float v = row[lane + 32 * t];
    s1 += v * asrc[lane + 32 * t];
    s2 += v * adst[lane + 32 * t];
  }
  s1 = wave_reduce_sum(s1);
  s2 = wave_reduce_sum(s2);
  if (lane == 0) { a_s[node] = s1; a_d[node] = s2; }
}

// ---------------------------------------------------------------------------
// GAT aggregate: softmax(leaky_relu(a_s[nbr]+a_d)) weighted message sum + bias
// wave per node; lanes 0..16 own one edge each; broadcast via shfl.
// ---------------------------------------------------------------------------
__global__ void __launch_bounds__(256) aggregate_kernel(const float* __restrict__ hp,
                                                        const int* __restrict__ nbr,
                                                        const float* __restrict__ a_s,
                                                        const float* __restrict__ a_d,
                                                        const float* __restrict__ bias,
                                                        float* __restrict__ hout,
                                                        int relu) {
  const int wave = threadIdx.x >> 5, lane = threadIdx.x & 31;
  const int gnode = blockIdx.x * 8 + wave;   // flat over B*N
  const int b = gnode >> 10;                 // N = 1024
  const int* nb = nbr + (size_t)gnode * KP1;
  const int nj = (lane < KP1) ? nb[lane] : 0;
  float e;
  if (lane < KP1) {
    const float v = a_s[(b << 10) + nj] + a_d[gnode];
    e = (v >= 0.f) ? v : NEG_SLOPE * v;      // leaky relu
  } else {
    e = -INFINITY;
  }
  const float emax = wave_reduce_max(e);
  const float ex = (lane < KP1) ? __expf(e - emax) : 0.f;
  const float esum = wave_reduce_sum(ex);
  const float att = ex / esum;

  float acc[D_ / 32];
#pragma unroll
  for (int t = 0; t < D_ / 32; ++t) acc[t] = 0.f;
  const float* hpb = hp + ((size_t)(b << 10)) * D_;
#pragma unroll 1
  for (int j = 0; j < KP1; ++j) {
    const float aj = __shfl(att, j, 32);
    const int idx = __shfl(nj, j, 32);
    const float* src = hpb + (size_t)idx * D_;
#pragma unroll
    for (int t = 0; t < D_ / 32; ++t) acc[t] += aj * src[lane + 32 * t];
  }
  float* dst = hout + (size_t)gnode * D_;
#pragma unroll
  for (int t = 0; t < D_ / 32; ++t) {
    float v = acc[t] + bias[lane + 32 * t];
    if (relu) v = fmaxf(v, 0.f);
    dst[lane + 32 * t] = v;
  }
}

// ---------------------------------------------------------------------------
// Global max pool over nodes. grid = B, block = D
// ---------------------------------------------------------------------------
__global__ void __launch_bounds__(D_) maxpool_kernel(const float* __restrict__ h,
                                                     float* __restrict__ pooled) {
  const int b = blockIdx.x, d = threadIdx.x;
  const float* hb = h + (size_t)b * N_ * D_;
  float m = -INFINITY;
  for (int n = 0; n < N_; ++n) m = fmaxf(m, hb[(size_t)n * D_ + d]);
  pooled[b * D_ + d] = m;
}

// ---------------------------------------------------------------------------
// out[b,c] = pooled[b] . Wc[:,c] + bc[c]. grid = B*2 waves
// ---------------------------------------------------------------------------
__global__ void __launch_bounds__(32) classifier_kernel(const float* __restrict__ pooled,
                                                        const float* __restrict__ Wc,
                                                        const float* __restrict__ bc,
                                                        float* __restrict__ out) {
  const int o = blockIdx.x;
  const int bI = o >> 1, cI = o & 1;
  const int lane = threadIdx.x;
  float s = 0.f;
#pragma unroll
  for (int t = 0; t < D_ / 32; ++t) {
    const int d = lane + 32 * t;
    s += pooled[bI * D_ + d] * Wc[d * 2 + cI];
  }
  s = wave_reduce_sum(s);
  if (lane == 0) out[o] = s + bc[cI];
}

// ---------------------------------------------------------------------------
extern "C" void kernel_launch(void* const* d_in, const int* in_sizes, int n_in,
                              void* d_out, int out_size, void* d_ws, size_t ws_size,
                              hipStream_t stream) {
  (void)in_sizes; (void)n_in; (void)out_size; (void)ws_size;
  const float* x = (const float*)d_in[0];
  const float* W[3]    = {(const float*)d_in[1], (const float*)d_in[5], (const float*)d_in[9]};
  const float* asrc[3] = {(const float*)d_in[2], (const float*)d_in[6], (const float*)d_in[10]};
  const float* adst[3] = {(const float*)d_in[3], (const float*)d_in[7], (const float*)d_in[11]};
  const float* bias[3] = {(const float*)d_in[4], (const float*)d_in[8], (const float*)d_in[12]};
  const float* Wc = (const float*)d_in[13];
  const float* bc = (const float*)d_in[14];
  float* out = (float*)d_out;

  char* ws = (char*)d_ws;
  const size_t HND = (size_t)B_ * N_ * D_ * sizeof(float);   // 32 MB
  float* hA     = (float*)(ws);                  // layer input / aggregate output
  float* hB     = (float*)(ws + HND);            // GEMM output hp
  float* sq     = (float*)(ws + 2 * HND);                    // B*N floats (64 KB)
  float* a_s    = (float*)(ws + 2 * HND + 64 * 1024);        // B*N floats
  float* a_d    = (float*)(ws + 2 * HND + 128 * 1024);       // B*N floats
  float* pooled = (float*)(ws + 2 * HND + 192 * 1024);       // B*D floats (32 KB)
  int*   nbr    = (int*)  (ws + 2 * HND + 256 * 1024);       // B*N*17 ints (~1.1 MB)

  normalize_kernel<<<B_, D_, 0, stream>>>(x, hA);
  sqnorm_kernel<<<B_ * N_ / 8, 256, 0, stream>>>(x, sq);
  knn_kernel<<<dim3(N_ / 16, B_), 32, 0, stream>>>(x, sq, nbr);

  for (int l = 0; l < 3; ++l) {
    gemm_kernel<<<dim3(N_ / 64, D_ / 64, B_), 128, 0, stream>>>(hA, W[l], hB);
    dots_kernel<<<B_ * N_ / 8, 256, 0, stream>>>(hB, asrc[l], adst[l], a_s, a_d);
    aggregate_kernel<<<B_ * N_ / 8, 256, 0, stream>>>(hB, nbr, a_s, a_d, bias[l], hA,
                                                      (l < 2) ? 1 : 0);
  }

  maxpool_kernel<<<B_, D_, 0, stream>>>(hA, pooled);
  classifier_kernel<<<B_ * 2, 32, 0, stream>>>(pooled, Wc, bc, out);
}